// Custom_Softmax_75393855914393
// MI455X (gfx1250) — compile-verified
//
#include <hip/hip_runtime.h>
#include <stdint.h>

#define S_LEN 2048
#define HEADS 16
#define WAVES_PER_BLOCK 4
#define THREADS (WAVES_PER_BLOCK * 32)
#define CHUNK 128   /* elements per wave per iteration: 32 lanes x 4 ints */

__device__ __forceinline__ int clamp_q(float v) {
    int q = (int)__builtin_rintf(v);          // round-half-even like jnp.round
    q = q > 127 ? 127 : q;
    q = q < -128 ? -128 : q;
    return q & 0xFF;
}

__global__ __launch_bounds__(THREADS) void softmax_quant_rows(
    const int* __restrict__ x_q,
    const float* __restrict__ scale_x,
    const float* __restrict__ scale_out,
    uint8_t* __restrict__ out_q,
    float* __restrict__ so_out)
{
    __shared__ int lds[WAVES_PER_BLOCK * S_LEN];   // 8 KB per wave (one full row)

    const int tid  = threadIdx.x;
    const int lane = tid & 31;
    // Force wave-uniform row index into SGPRs: scalar loop bounds, scalar
    // scale loads (s_load/KMcnt), and a clean SGPR-pair SADDR for the async op.
    const int wave = __builtin_amdgcn_readfirstlane(tid >> 5);
    const int row  = blockIdx.x * WAVES_PER_BLOCK + wave;   // 0 .. H*S-1 (scalar)
    const int s    = row & (S_LEN - 1);
    const int len  = s + 1;                                  // causal row length
    const int nIter = (len + CHUNK - 1) / CHUNK;             // <= 16 (scalar)

    const float sx = scale_x[row];     // (H, 2048) flattened == row index
    const float so = scale_out[row];

    int* slice = &lds[wave * S_LEN];

    // ---------- async HBM -> LDS stage of the valid row prefix (512B chunks) ----------
    const uint32_t lds_base = (uint32_t)(uintptr_t)slice;          // low 32 bits = LDS offset
    const uint64_t gbase    = (uint64_t)(uintptr_t)(x_q + (size_t)row * S_LEN); // SGPR pair
    for (int it = 0; it < nIter; ++it) {
        uint32_t boff = (uint32_t)it * 512u + (uint32_t)lane * 16u;
        uint32_t ldsa = lds_base + boff;
        asm volatile("global_load_async_to_lds_b128 %0, %1, %2"
                     :: "v"(ldsa), "v"(boff), "s"(gbase) : "memory");
    }
    asm volatile("s_wait_asynccnt 0" ::: "memory");

    // ---------- pass 1: row max in the int domain (sx > 0, so max commutes) ----------
    int imax = (int)0x80000000;
    for (int it = 0; it < nIter; ++it) {
        int idx = it * CHUNK + lane * 4;
        int4 v = *(const int4*)(slice + idx);
        if (idx + 0 < len && v.x > imax) imax = v.x;
        if (idx + 1 < len && v.y > imax) imax = v.y;
        if (idx + 2 < len && v.z > imax) imax = v.z;
        if (idx + 3 < len && v.w > imax) imax = v.w;
    }
    for (int off = 16; off; off >>= 1) {
        int o = __shfl_xor(imax, off, 32);
        if (o > imax) imax = o;
    }
    const float m = sx * (float)imax;

    // ---------- pass 2: exp + sum, overwrite LDS in place with f32 exp values ----------
    float sum = 0.0f;
    for (int it = 0; it < nIter; ++it) {
        int idx = it * CHUNK + lane * 4;
        int4 v = *(const int4*)(slice + idx);
        float4 e;
        e.x = (idx + 0 < len) ? __expf(sx * (float)v.x - m) : 0.0f;
        e.y = (idx + 1 < len) ? __expf(sx * (float)v.y - m) : 0.0f;
        e.z = (idx + 2 < len) ? __expf(sx * (float)v.z - m) : 0.0f;
        e.w = (idx + 3 < len) ? __expf(sx * (float)v.w - m) : 0.0f;
        sum += (e.x + e.y) + (e.z + e.w);
        *(float4*)(void*)(slice + idx) = e;
    }
    for (int off = 16; off; off >>= 1) sum += __shfl_xor(sum, off, 32);
    const float recip = 1.0f / (sum * so);   // out = rint(e/sum/so) = rint(e*recip)

    // ---------- pass 3a: quantize valid chunks, pack 4x int8, NT coalesced stores ----------
    uint32_t* outRow = (uint32_t*)(out_q + (size_t)row * S_LEN);
    for (int it = 0; it < nIter; ++it) {
        int idx = it * CHUNK + lane * 4;
        float4 e = *(const float4*)(const void*)(slice + idx);   // 0.0f at masked tail
        uint32_t packed = (uint32_t)clamp_q(e.x * recip)
                        | ((uint32_t)clamp_q(e.y * recip) << 8)
                        | ((uint32_t)clamp_q(e.z * recip) << 16)
                        | ((uint32_t)clamp_q(e.w * recip) << 24);
        __builtin_nontemporal_store(packed, &outRow[idx >> 2]);
    }
    // ---------- pass 3b: zero-fill above the staged region (upper triangle) ----------
    for (int it = nIter; it < S_LEN / CHUNK; ++it) {
        int idx = it * CHUNK + lane * 4;
        __builtin_nontemporal_store(0u, &outRow[idx >> 2]);
    }

    if (lane == 0) so_out[row] = so;   // second tuple output
}

extern "C" void kernel_launch(void* const* d_in, const int* in_sizes, int n_in,
                              void* d_out, int out_size, void* d_ws, size_t ws_size,
                              hipStream_t stream) {
    const int*   x_q       = (const int*)d_in[0];
    const float* scale_x   = (const float*)d_in[1];
    const float* scale_out = (const float*)d_in[2];

    uint8_t* out_q  = (uint8_t*)d_out;                                   // int8 (B,H,S,S)
    float*   so_out = (float*)(out_q + (size_t)HEADS * S_LEN * S_LEN);   // float32 (H,S)

    const int rows = HEADS * S_LEN;                  // 32768 rows, 1 wave each
    dim3 grid(rows / WAVES_PER_BLOCK);
    softmax_quant_rows<<<grid, THREADS, 0, stream>>>(x_q, scale_x, scale_out, out_q, so_out);
}